// TreeLSTM_10033043603924
// MI455X (gfx1250) — compile-verified
//
#include <hip/hip_runtime.h>
#include <cmath>

// ---------------------------------------------------------------------------
// TreeLSTM (SPINN shift-reduce) for MI455X / gfx1250, wave32 + fp32 WMMA.
//
//  Prep:    zero stack + sync counter, build transposed weights (WpT, WgT,
//           WrT) so B-fragments load as contiguous float2 (global_load_b64).
//  Phase 1: buffers[b,t,0:D]=h, [D:2D]=c from two 32768x256x256 GEMMs
//           (V_WMMA_F32_16X16X4_F32) + fused sigmoid/tanh gating.
//  Phase 2: one persistent kernel (32 blocks), agent-scope grid barrier
//           between the 255 stack-machine steps; each reduce = 256 waves,
//           one 16x16 h/c tile per wave with 5 gate accumulators, K=512.
// ---------------------------------------------------------------------------

#define B_      256
#define T_      128
#define E_      256
#define D_      256
#define TWO_D   512
#define FIVE_D  1280
#define NSTEP   (2 * T_ - 1)
#define SMAX    T_            // stack depth, matches reference (B, T, 2D)
#define NB_SEQ  32            // blocks in persistent sequential kernel
#define NT_SEQ  256           // threads per block (8 wave32)

typedef float v2f __attribute__((ext_vector_type(2)));
typedef float v8f __attribute__((ext_vector_type(8)));

__device__ __forceinline__ float sigm(float x) {
  return 1.0f / (1.0f + __expf(-x));
}

// Agent-scope grid barrier: monotonic counter in workspace (zeroed per call).
__device__ __forceinline__ void grid_sync(unsigned* cnt, unsigned gen) {
  __builtin_amdgcn_fence(__ATOMIC_RELEASE, "agent");
  __syncthreads();
  if (threadIdx.x == 0) {
    __hip_atomic_fetch_add(cnt, 1u, __ATOMIC_ACQ_REL, __HIP_MEMORY_SCOPE_AGENT);
    while (__hip_atomic_load(cnt, __ATOMIC_RELAXED, __HIP_MEMORY_SCOPE_AGENT) <
           gen * (unsigned)NB_SEQ) {
      __builtin_amdgcn_s_sleep(1);
    }
  }
  __syncthreads();
  __builtin_amdgcn_fence(__ATOMIC_ACQUIRE, "agent");
}

// ---------------------------------------------------------------------------
// Prep: zero stack + counter, and transpose Wp, Wg, Wr into workspace.
//   WpT[c*E + k] = Wp[k*D + c]      (D x E)
//   WgT[c*E + k] = Wg[k*D + c]      (D x E)
//   WrT[c*2D + k] = Wr[k*5D + c]    (5D x 2D)
// ---------------------------------------------------------------------------
#define NZ_STACK ((size_t)B_ * SMAX * TWO_D)
#define N_WT     ((size_t)E_ * D_)
#define N_WRT    ((size_t)TWO_D * FIVE_D)

__global__ void tl_prep_kernel(const float* __restrict__ Wp,
                               const float* __restrict__ Wg,
                               const float* __restrict__ Wr,
                               float* __restrict__ stack,
                               float* __restrict__ WpT,
                               float* __restrict__ WgT,
                               float* __restrict__ WrT,
                               unsigned* __restrict__ cnt) {
  const size_t total = NZ_STACK + 2 * N_WT + N_WRT;
  size_t i = (size_t)blockIdx.x * blockDim.x + threadIdx.x;
  if (i == 0) *cnt = 0u;
  for (; i < total; i += (size_t)gridDim.x * blockDim.x) {
    if (i < NZ_STACK) {
      stack[i] = 0.0f;
    } else if (i < NZ_STACK + N_WT) {
      const size_t j = i - NZ_STACK;
      const size_t c = j / E_, k = j % E_;
      WpT[j] = Wp[k * D_ + c];
    } else if (i < NZ_STACK + 2 * N_WT) {
      const size_t j = i - NZ_STACK - N_WT;
      const size_t c = j / E_, k = j % E_;
      WgT[j] = Wg[k * D_ + c];
    } else {
      const size_t j = i - NZ_STACK - 2 * N_WT;
      const size_t c = j / TWO_D, k = j % TWO_D;
      WrT[j] = Wr[k * FIVE_D + c];
    }
  }
}

// ---------------------------------------------------------------------------
// Phase 1: buffers = [sigmoid(xWg+bg)*tanh(xWp+bp) , xWp+bp]
// One wave per 16x16 output tile; 2 fp32-WMMA accumulators (proj & gate).
// ---------------------------------------------------------------------------
__global__ __launch_bounds__(256) void tl_preproj_kernel(
    const float* __restrict__ x, const float* __restrict__ WpT,
    const float* __restrict__ bp, const float* __restrict__ WgT,
    const float* __restrict__ bg, float* __restrict__ buffers) {
  const int lane   = threadIdx.x & 31;
  const int wid    = (int)((blockIdx.x * blockDim.x + threadIdx.x) >> 5); // 0..32767
  const int tile_m = wid >> 4;                 // 0..2047 rows of (B*T)
  const int tile_n = wid & 15;                 // 0..15 cols of D
  const int row0   = tile_m * 16;
  const int arow   = lane & 15;
  const int koff   = (lane >> 4) << 1;         // 0 or 2
  const int bcol   = tile_n * 16 + (lane & 15);

  v8f accP = {0.f, 0.f, 0.f, 0.f, 0.f, 0.f, 0.f, 0.f};
  v8f accG = {0.f, 0.f, 0.f, 0.f, 0.f, 0.f, 0.f, 0.f};

  const float* xrow = x + (size_t)(row0 + arow) * E_;
  const float* wp   = WpT + (size_t)bcol * E_;   // contiguous in K
  const float* wg   = WgT + (size_t)bcol * E_;
#pragma unroll 4
  for (int k0 = 0; k0 < E_; k0 += 4) {
    const int kk = k0 + koff;
    v2f a  = *(const v2f*)(xrow + kk);           // A[row][kk, kk+1]
    v2f bP = *(const v2f*)(wp + kk);             // WpT[col][kk, kk+1]
    v2f bG = *(const v2f*)(wg + kk);
    accP = __builtin_amdgcn_wmma_f32_16x16x4_f32(false, a, false, bP, (short)0,
                                                 accP, false, false);
    accG = __builtin_amdgcn_wmma_f32_16x16x4_f32(false, a, false, bG, (short)0,
                                                 accG, false, false);
  }

  const float bpv  = bp[bcol];
  const float bgv  = bg[bcol];
  const int  rbase = row0 + ((lane >> 4) << 3);
#pragma unroll
  for (int v = 0; v < 8; ++v) {
    const int r   = rbase + v;
    const float c = accP[v] + bpv;
    const float h = sigm(accG[v] + bgv) * tanhf(c);
    buffers[(size_t)r * TWO_D + bcol]      = h;
    buffers[(size_t)r * TWO_D + D_ + bcol] = c;
  }
}

// ---------------------------------------------------------------------------
// Phase 2: persistent stack-machine kernel. transitions are uniform across
// the batch, so every block tracks ptr/bptr identically in registers.
// ---------------------------------------------------------------------------
__global__ __launch_bounds__(NT_SEQ) void tl_seq_kernel(
    const float* __restrict__ buffers, const float* __restrict__ WrT,
    const float* __restrict__ br, const int* __restrict__ transitions,
    float* __restrict__ stack, float* __restrict__ out,
    unsigned* __restrict__ cnt) {
  const int tid    = (int)(blockIdx.x * blockDim.x + threadIdx.x); // 0..8191
  const int lane   = threadIdx.x & 31;
  const int wid    = tid >> 5;           // 0..255 waves
  const int tile_m = wid >> 4;           // batch-row tile 0..15
  const int tile_n = wid & 15;           // D-col tile 0..15

  // Step-invariant per-wave constants (hoisted out of the 255-step loop).
  const int row0  = tile_m * 16;
  const int arow  = lane & 15;
  const int koff  = (lane >> 4) << 1;
  const int ncol  = lane & 15;
  const int rowb  = row0 + arow;
  const int colD  = tile_n * 16 + ncol;
  const int rbase = row0 + ((lane >> 4) << 3);

  const float* wb[5];   // WrT column bases per gate (contiguous in K)
  float brv[5];         // gate biases
#pragma unroll
  for (int g = 0; g < 5; ++g) {
    wb[g]  = WrT + (size_t)(g * D_ + colD) * TWO_D;
    brv[g] = br[g * D_ + colD];
  }

  unsigned gen = 0;
  int ptr = 0, bptr = T_ - 1;

  for (int t = 0; t < NSTEP; ++t) {
    const int tr = transitions[(size_t)t * B_];   // uniform across batch
    if (tr == 0) {
      // ---- SHIFT: stack[:, ptr] = buffers[:, bptr] ----
      for (int i = tid; i < B_ * TWO_D; i += NB_SEQ * NT_SEQ) {
        const int b = i >> 9;             // /512
        const int c = i & (TWO_D - 1);
        stack[((size_t)b * SMAX + ptr) * TWO_D + c] =
            buffers[((size_t)b * T_ + bptr) * TWO_D + c];
      }
      ptr += 1;
      bptr -= 1;
    } else {
      // ---- REDUCE: [h_l,h_r] @ Wr + br -> gates -> (h,c) at stack[dst] ----
      const int l   = (ptr - 2 > 0) ? ptr - 2 : 0;
      const int r   = (ptr - 1 > 0) ? ptr - 1 : 0;
      const int dst = l;

      const float* Al = stack + ((size_t)rowb * SMAX + l) * TWO_D; // h_l row
      const float* Ar = stack + ((size_t)rowb * SMAX + r) * TWO_D; // h_r row

      v8f acc[5];
#pragma unroll
      for (int g = 0; g < 5; ++g)
        acc[g] = (v8f){0.f, 0.f, 0.f, 0.f, 0.f, 0.f, 0.f, 0.f};

      // K = 0..255 : left child h
      for (int k0 = 0; k0 < D_; k0 += 4) {
        const int kk = k0 + koff;
        v2f a = *(const v2f*)(Al + kk);
#pragma unroll
        for (int g = 0; g < 5; ++g) {
          v2f b = *(const v2f*)(wb[g] + kk);
          acc[g] = __builtin_amdgcn_wmma_f32_16x16x4_f32(
              false, a, false, b, (short)0, acc[g], false, false);
        }
      }
      // K = 256..511 : right child h
      for (int k0 = 0; k0 < D_; k0 += 4) {
        const int kk = k0 + koff;
        v2f a = *(const v2f*)(Ar + kk);
#pragma unroll
        for (int g = 0; g < 5; ++g) {
          v2f b = *(const v2f*)(wb[g] + D_ + kk);
          acc[g] = __builtin_amdgcn_wmma_f32_16x16x4_f32(
              false, a, false, b, (short)0, acc[g], false, false);
        }
      }

      // Gate epilogue into registers (reads of c_l/c_r BEFORE overwrite).
      float hout[8], cout[8];
#pragma unroll
      for (int v = 0; v < 8; ++v) {
        const int rb   = rbase + v;
        const float iv = sigm(acc[0][v] + brv[0]);
        const float fl = sigm(acc[1][v] + brv[1]);
        const float fr = sigm(acc[2][v] + brv[2]);
        const float gv = tanhf(acc[3][v] + brv[3]);
        const float ov = sigm(acc[4][v] + brv[4]);
        const float cl = stack[((size_t)rb * SMAX + l) * TWO_D + D_ + colD];
        const float cr = stack[((size_t)rb * SMAX + r) * TWO_D + D_ + colD];
        const float cc = iv * gv + fl * cl + fr * cr;
        cout[v] = cc;
        hout[v] = ov * tanhf(cc);
      }

      // stack[dst] aliases stack[l]: all waves must finish reading first.
      grid_sync(cnt, ++gen);

#pragma unroll
      for (int v = 0; v < 8; ++v) {
        const int rb = rbase + v;
        stack[((size_t)rb * SMAX + dst) * TWO_D + colD]      = hout[v];
        stack[((size_t)rb * SMAX + dst) * TWO_D + D_ + colD] = cout[v];
      }
      ptr -= 1;
    }
    grid_sync(cnt, ++gen);   // step boundary
  }

  // Final output: h part of stack top.
  const int os = (ptr - 1 > 0) ? ptr - 1 : 0;
  for (int i = tid; i < B_ * D_; i += NB_SEQ * NT_SEQ) {
    const int b = i >> 8;
    const int c = i & (D_ - 1);
    out[i] = stack[((size_t)b * SMAX + os) * TWO_D + c];
  }
}

// ---------------------------------------------------------------------------
extern "C" void kernel_launch(void* const* d_in, const int* in_sizes, int n_in,
                              void* d_out, int out_size, void* d_ws,
                              size_t ws_size, hipStream_t stream) {
  (void)in_sizes; (void)n_in; (void)out_size; (void)ws_size;

  const float* x   = (const float*)d_in[0];
  const float* Wp  = (const float*)d_in[1];
  const float* bp  = (const float*)d_in[2];
  const float* Wg  = (const float*)d_in[3];
  const float* bg  = (const float*)d_in[4];
  const float* Wr  = (const float*)d_in[5];
  const float* br  = (const float*)d_in[6];
  const int* trans = (const int*)d_in[7];
  float* out = (float*)d_out;

  // Workspace: buffers (64MB) | stack (64MB) | WpT | WgT | WrT | counter.
  float* buffers = (float*)d_ws;
  float* stack   = buffers + (size_t)B_ * T_ * TWO_D;
  float* WpT     = stack + NZ_STACK;
  float* WgT     = WpT + N_WT;
  float* WrT     = WgT + N_WT;
  unsigned* cnt  = (unsigned*)(WrT + N_WRT);

  tl_prep_kernel<<<1024, 256, 0, stream>>>(Wp, Wg, Wr, stack, WpT, WgT, WrT,
                                           cnt);
  // 32768 tiles of 16x16, 8 waves/block -> 4096 blocks.
  tl_preproj_kernel<<<4096, 256, 0, stream>>>(x, WpT, bp, WgT, bg, buffers);
  // Persistent sequential kernel: 32 blocks co-resident, grid-barrier synced.
  tl_seq_kernel<<<NB_SEQ, NT_SEQ, 0, stream>>>(buffers, WrT, br, trans, stack,
                                               out, cnt);
}